// FOFE_Encoding_7146825580657
// MI455X (gfx1250) — compile-verified
//
#include <hip/hip_runtime.h>
#include <hip/hip_bf16.h>
#include <stdint.h>

// FOFE encoding, MI455X / gfx1250.
// B=256, S=256, W=20, V=512. Output: [B*S*512] f32 histogram + [B] lengths.
// Roofline: ~134 MB written + ~5 MB read => ~6us floor @ 23.3 TB/s; compute
// is 1.3M scatter-adds => purely store-bandwidth bound, no matmul structure
// (the one-hot operand varies per row), so WMMA is inapplicable. CDNA5 paths
// used instead: wave32-private LDS tiles + ds_add_f32 scatter, and
// GLOBAL_STORE_ASYNC_FROM_LDS_B128 (ASYNCcnt) for the 134MB writeback with
// no VGPR round-trip.

#define B_  256
#define S_  256
#define W_  20
#define V_  512
#define WAVES_PER_BLOCK 8
#define THREADS_ (WAVES_PER_BLOCK * 32)

typedef __attribute__((address_space(3))) float lds_f32;

__global__ __launch_bounds__(THREADS_) void fofe_rows_kernel(
    const int* __restrict__ sents,      // [B*S, W] int32 in [0, V)
    const float* __restrict__ ff,       // [1] forgetting factor alpha
    float* __restrict__ out)            // [B*S, V] f32
{
    __shared__ float lds[WAVES_PER_BLOCK * V_];

    const int lane  = threadIdx.x & 31;
    const int wave  = threadIdx.x >> 5;
    const int row   = blockIdx.x * WAVES_PER_BLOCK + wave;   // [0, B*S)
    const int wbase = wave * V_;

    // gfx1250 prefetch of this wave's index row (global_prefetch_b8).
    __builtin_prefetch(&sents[(size_t)row * W_], 0, 3);

    // 1) Zero this wave's private 2KB LDS row: 4x ds_store_b128 per lane.
    //    Chunk layout (j*32+lane)*4 floats -> 16B-aligned, conflict-free.
    const float4 z4 = make_float4(0.f, 0.f, 0.f, 0.f);
#pragma unroll
    for (int j = 0; j < 4; ++j) {
        *(float4*)&lds[wbase + (j * 32 + lane) * 4] = z4;
    }

    // 2) Scatter: lanes 0..19 each own one word position k = lane, weight
    //    alpha^(W-1-k) = exp2((W-1-k) * log2(alpha)) via the hardware
    //    transcendentals (v_log_f32 + v_mul_f32 + v_exp_f32), no libcall.
    //    ds_add_f32 atomics absorb duplicate vocab ids; same-wave DS ops are
    //    in-order so the zero-fill above is already visible.
    const float alpha = ff[0];
    if (lane < W_) {
        const float wgt = __builtin_amdgcn_exp2f(
            (float)(W_ - 1 - lane) * __builtin_amdgcn_logf(alpha));
        const int c = sents[(size_t)row * W_ + lane];
        atomicAdd(&lds[wbase + c], wgt);                     // ds_add_f32
    }

    // 3) Writeback: async LDS -> global DMA (GLOBAL_STORE_ASYNC_FROM_LDS_B128,
    //    ASYNCcnt). 16B per lane per instruction = one coalesced 512B store
    //    per wave per issue; the 134MB payload never touches VGPRs.
    float* gout = out + (size_t)row * V_;
    asm volatile("s_wait_dscnt 0x0" ::: "memory");           // atomic landed in LDS
#pragma unroll
    for (int j = 0; j < 4; ++j) {
        const int e = (j * 32 + lane) * 4;                   // float index in row
        uint64_t ga = (uint64_t)(uintptr_t)(gout + e);
        uint32_t la = (uint32_t)(uintptr_t)(lds_f32*)&lds[wbase + e];
        asm volatile("global_store_async_from_lds_b128 %0, %1, off"
                     :: "v"(ga), "v"(la)
                     : "memory");
    }
    asm volatile("s_wait_asynccnt 0x0" ::: "memory");        // drain before wave exit
}

__global__ void fofe_lengths_kernel(const int* __restrict__ lengths,
                                    float* __restrict__ out_tail, int n)
{
    const int i = blockIdx.x * blockDim.x + threadIdx.x;
    if (i < n) out_tail[i] = (float)lengths[i];
}

extern "C" void kernel_launch(void* const* d_in, const int* in_sizes, int n_in,
                              void* d_out, int out_size, void* d_ws, size_t ws_size,
                              hipStream_t stream) {
    (void)in_sizes; (void)n_in; (void)d_ws; (void)ws_size; (void)out_size;

    const int*   sents   = (const int*)d_in[0];    // [B,S,W] int32
    const int*   lengths = (const int*)d_in[1];    // [B] int32
    const float* ff      = (const float*)d_in[2];  // [1] float32
    float*       out     = (float*)d_out;

    const int rows = B_ * S_;                      // 65536 rows, 1 wave each
    fofe_rows_kernel<<<dim3(rows / WAVES_PER_BLOCK), dim3(THREADS_), 0, stream>>>(
        sents, ff, out);

    float* tail = out + (size_t)rows * V_;
    fofe_lengths_kernel<<<dim3((B_ + 255) / 256), dim3(256), 0, stream>>>(
        lengths, tail, B_);
}